// MonocularCameras_30150670418677
// MI455X (gfx1250) — compile-verified
//
#include <hip/hip_runtime.h>
#include <math.h>

// Inclusive scan of T=2^20 SE(3) matrices under (non-commutative) matmul.
// 3-phase scan; cross-thread 4x4 matmul batches are done with
// V_WMMA_F32_16X16X4_F32 (4 independent 4x4 products per WMMA, diagonal blocks).

typedef float v2f __attribute__((ext_vector_type(2)));
typedef float v8f __attribute__((ext_vector_type(8)));

#define NTHREADS 256
#define NBLOCKS  256
#define MS       16   // floats per 4x4 matrix slot in LDS

struct Aff { float r[9]; float t[3]; };

__device__ __forceinline__ Aff aff_identity() {
    Aff a;
    a.r[0]=1.f; a.r[1]=0.f; a.r[2]=0.f;
    a.r[3]=0.f; a.r[4]=1.f; a.r[5]=0.f;
    a.r[6]=0.f; a.r[7]=0.f; a.r[8]=1.f;
    a.t[0]=0.f; a.t[1]=0.f; a.t[2]=0.f;
    return a;
}

// o = A compose B  (A is the earlier/left matrix:  o = A @ B)
__device__ __forceinline__ Aff aff_mul(const Aff& A, const Aff& B) {
    Aff o;
#pragma unroll
    for (int i = 0; i < 3; ++i) {
        const float a0 = A.r[3*i+0], a1 = A.r[3*i+1], a2 = A.r[3*i+2];
#pragma unroll
        for (int j = 0; j < 3; ++j)
            o.r[3*i+j] = a0*B.r[j] + a1*B.r[3+j] + a2*B.r[6+j];
        o.t[i] = a0*B.t[0] + a1*B.t[1] + a2*B.t[2] + A.t[i];
    }
    return o;
}

__device__ __forceinline__ Aff make_dT(const float* __restrict__ qp,
                                       const float* __restrict__ tp, long idx) {
    const float w = qp[idx*4+0], x = qp[idx*4+1], y = qp[idx*4+2], z = qp[idx*4+3];
    const float n   = sqrtf(w*w + x*x + y*y + z*z);
    const float inv = 1.f / fmaxf(n, 1e-12f);
    const float r = w*inv, i = x*inv, j = y*inv, k = z*inv;
    const float two_s = 2.f / (r*r + i*i + j*j + k*k);
    Aff o;
    o.r[0] = 1.f - two_s*(j*j + k*k);
    o.r[1] = two_s*(i*j - k*r);
    o.r[2] = two_s*(i*k + j*r);
    o.r[3] = two_s*(i*j + k*r);
    o.r[4] = 1.f - two_s*(i*i + k*k);
    o.r[5] = two_s*(j*k - i*r);
    o.r[6] = two_s*(i*k - j*r);
    o.r[7] = two_s*(j*k + i*r);
    o.r[8] = 1.f - two_s*(i*i + j*j);
    o.t[0] = tp[idx*3+0]; o.t[1] = tp[idx*3+1]; o.t[2] = tp[idx*3+2];
    return o;
}

__device__ __forceinline__ void aff_to_mat16(const Aff& a, float* m) {
#pragma unroll
    for (int i = 0; i < 3; ++i) {
        m[4*i+0]=a.r[3*i+0]; m[4*i+1]=a.r[3*i+1]; m[4*i+2]=a.r[3*i+2]; m[4*i+3]=a.t[i];
    }
    m[12]=0.f; m[13]=0.f; m[14]=0.f; m[15]=1.f;
}

__device__ __forceinline__ Aff mat16_to_aff(const float* m) {
    Aff a;
#pragma unroll
    for (int i = 0; i < 3; ++i) {
        a.r[3*i+0]=m[4*i+0]; a.r[3*i+1]=m[4*i+1]; a.r[3*i+2]=m[4*i+2]; a.t[i]=m[4*i+3];
    }
    return a;
}

// Hillis-Steele inclusive scan of 256 4x4 matrices in LDS (ping-pong b0/b1).
// Combine step uses V_WMMA_F32_16X16X4_F32: 4 stacked left matrices (A 16x4)
// x 4 concatenated right matrices (B 4x16); diagonal 4x4 blocks of C are the
// 4 pair-products. Entries j<d multiply against identity -> wave-uniform
// control flow around the WMMA (EXEC all-ones requirement).
__device__ float* lds_scan256_wmma(float* b0, float* b1, const float* ident, int tid) {
    float* src = b0;
    float* dst = b1;
    const int  lane  = tid & 31;
    const int  wave  = tid >> 5;
    const int  m     = lane & 15;          // row index of A / col index of B
    const int  kh    = lane >> 4;          // K half (0: K=0,1  1: K=2,3)
    const int  g     = m >> 2;             // pair-group within the WMMA
    const int  rc    = m & 3;              // row within group (A) / col within group (B)
    const bool valid = ((m >> 3) == kh);   // lanes that own diagonal-block columns
    const bool hi    = ((g & 1) != 0);     // this lane's block sits in C VGPRs 4..7

    for (int d = 1; d < 256; d <<= 1) {
#pragma unroll
        for (int gi = 0; gi < 8; ++gi) {
            const int G  = wave * 8 + gi;  // 8 waves x 8 groups = 64 groups = 256 pairs
            const int jj = 4 * G + g;
            const float* Lp = (jj >= d) ? (src + (size_t)(jj - d) * MS) : ident;
            const float* Rp = src + (size_t)jj * MS;
            v2f a, b;
            a.x = Lp[rc*4 + 2*kh];         // A[m][2kh],   A 16x4 layout (ISA 7.12.2)
            a.y = Lp[rc*4 + 2*kh + 1];     // A[m][2kh+1]
            b.x = Rp[(2*kh + 0)*4 + rc];   // B[2kh  ][n]
            b.y = Rp[(2*kh + 1)*4 + rc];   // B[2kh+1][n]
            v8f cz = {0.f,0.f,0.f,0.f,0.f,0.f,0.f,0.f};
            v8f cd = __builtin_amdgcn_wmma_f32_16x16x4_f32(
                false, a, false, b, (short)0, cz, false, false);
            // Branchless half-select (4 v_cndmask) instead of dynamic extract.
            const float o0 = hi ? cd[4] : cd[0];
            const float o1 = hi ? cd[5] : cd[1];
            const float o2 = hi ? cd[6] : cd[2];
            const float o3 = hi ? cd[7] : cd[3];
            if (valid) {
                float* Op = dst + (size_t)jj * MS;
                Op[0*4+rc] = o0;
                Op[1*4+rc] = o1;
                Op[2*4+rc] = o2;
                Op[3*4+rc] = o3;
            }
        }
        __syncthreads();
        float* tmp = src; src = dst; dst = tmp;
    }
    return src;  // 8 steps -> result back in b0
}

__device__ __forceinline__ void init_ident(float* ident, int tid) {
    if (tid < MS) ident[tid] = ((tid & 3) == (tid >> 2)) ? 1.f : 0.f;
}

__global__ void k_totals(const float* __restrict__ q, const float* __restrict__ t,
                         float* __restrict__ blockTotals, int T, int seg) {
    __shared__ float b0[NTHREADS*MS];
    __shared__ float b1[NTHREADS*MS];
    __shared__ float ident[MS];
    const int tid = threadIdx.x;
    init_ident(ident, tid);
    const long base = ((long)blockIdx.x * NTHREADS + tid) * seg;
    Aff acc = aff_identity();
    for (int s = 0; s < seg; ++s) {
        const long idx = base + s;
        if (idx < T) acc = aff_mul(acc, make_dT(q, t, idx));
    }
    aff_to_mat16(acc, b0 + tid*MS);
    __syncthreads();
    float* res = lds_scan256_wmma(b0, b1, ident, tid);
    if (tid < MS) blockTotals[blockIdx.x*MS + tid] = res[255*MS + tid];
}

__global__ void k_scan_blocks(const float* __restrict__ blockTotals,
                              float* __restrict__ blockPrefix) {
    __shared__ float b0[NTHREADS*MS];
    __shared__ float b1[NTHREADS*MS];
    __shared__ float ident[MS];
    const int tid = threadIdx.x;
    init_ident(ident, tid);
#pragma unroll
    for (int f = 0; f < MS; ++f) b0[tid*MS + f] = blockTotals[tid*MS + f];
    __syncthreads();
    float* res = lds_scan256_wmma(b0, b1, ident, tid);
#pragma unroll
    for (int f = 0; f < MS; ++f)   // exclusive prefix: identity for block 0
        blockPrefix[tid*MS + f] = (tid == 0) ? ident[f] : res[(tid-1)*MS + f];
}

__global__ void k_apply(const float* __restrict__ q, const float* __restrict__ t,
                        const float* __restrict__ blockPrefix, float* __restrict__ out,
                        int T, int seg) {
    __shared__ float b0[NTHREADS*MS];
    __shared__ float b1[NTHREADS*MS];
    __shared__ float ident[MS];
    const int tid = threadIdx.x;
    init_ident(ident, tid);
    const long base = ((long)blockIdx.x * NTHREADS + tid) * seg;
    Aff acc = aff_identity();
    for (int s = 0; s < seg; ++s) {
        const long idx = base + s;
        if (idx < T) acc = aff_mul(acc, make_dT(q, t, idx));
    }
    aff_to_mat16(acc, b0 + tid*MS);
    __syncthreads();
    float* res = lds_scan256_wmma(b0, b1, ident, tid);

    const Aff P = mat16_to_aff(blockPrefix + blockIdx.x*MS);
    Aff cur = (tid > 0) ? aff_mul(P, mat16_to_aff(res + (tid-1)*MS)) : P;

    for (int s = 0; s < seg; ++s) {
        const long idx = base + s;
        if (idx < T) {
            cur = aff_mul(cur, make_dT(q, t, idx));   // cur == out[idx]
            float mmat[16];
            aff_to_mat16(cur, mmat);
            float4* op = (float4*)(out + idx * 16);
            op[0] = make_float4(mmat[0],  mmat[1],  mmat[2],  mmat[3]);
            op[1] = make_float4(mmat[4],  mmat[5],  mmat[6],  mmat[7]);
            op[2] = make_float4(mmat[8],  mmat[9],  mmat[10], mmat[11]);
            op[3] = make_float4(mmat[12], mmat[13], mmat[14], mmat[15]);
        }
    }
}

extern "C" void kernel_launch(void* const* d_in, const int* in_sizes, int n_in,
                              void* d_out, int out_size, void* d_ws, size_t ws_size,
                              hipStream_t stream) {
    const float* q   = (const float*)d_in[0];   // (T,4) f32
    const float* t   = (const float*)d_in[1];   // (T,3) f32
    float*       out = (float*)d_out;           // (T,4,4) f32
    const int T   = in_sizes[0] / 4;
    const int seg = (T + NBLOCKS*NTHREADS - 1) / (NBLOCKS*NTHREADS);

    float* blockTotals = (float*)d_ws;                 // 256 * 16 floats (16 KB)
    float* blockPrefix = blockTotals + NBLOCKS * MS;   // 256 * 16 floats (16 KB)

    k_totals     <<<NBLOCKS, NTHREADS, 0, stream>>>(q, t, blockTotals, T, seg);
    k_scan_blocks<<<1,       NTHREADS, 0, stream>>>(blockTotals, blockPrefix);
    k_apply      <<<NBLOCKS, NTHREADS, 0, stream>>>(q, t, blockPrefix, out, T, seg);
}